// Loss_42082089566555
// MI455X (gfx1250) — compile-verified
//
#include <hip/hip_runtime.h>
#include <cstdint>
#include <cstddef>

#define DIM   192
#define PLANE (DIM * DIM)         // 36864
#define TW    32
#define TH    16
#define NTHR  (TW * TH)           // 512
#define CHUNK 48
#define NCHUNK (DIM / CHUNK)      // 4
#define GX (DIM / TW)             // 6
#define GY (DIM / TH)             // 12
#define GZ (NCHUNK * 2)           // 8 (chunks x batch)
#define NBLK (GX * GY * GZ)       // 576
#define LH (TH + 2)               // 18 rows incl. halo
#define LSTRIDE 40                // col 3 = left halo, 4..35 interior, 36 = right halo
#define LPLANE (LH * LSTRIDE)     // 720 floats
#define NINT  (LH * 8)            // 144 b128 interior slots per array
#define NHALO (LH * 2)            // 36 b32 halo slots per array
#define NTOT (2.0 * 192.0 * 192.0 * 192.0)

// ---- CDNA5 async global->LDS (ASYNCcnt path) -------------------------------
__device__ __forceinline__ void async_ld_b32(uint32_t lds_byte_addr,
                                             const float* gaddr) {
  asm volatile("global_load_async_to_lds_b32 %0, %1, off"
               :: "v"(lds_byte_addr),
                  "v"((unsigned long long)(uintptr_t)gaddr)
               : "memory");
}
__device__ __forceinline__ void async_ld_b128(uint32_t lds_byte_addr,
                                              const float* gaddr) {
  asm volatile("global_load_async_to_lds_b128 %0, %1, off"
               :: "v"(lds_byte_addr),
                  "v"((unsigned long long)(uintptr_t)gaddr)
               : "memory");
}
__device__ __forceinline__ void wait_async0() {
#if __has_builtin(__builtin_amdgcn_s_wait_asynccnt)
  __builtin_amdgcn_s_wait_asynccnt(0);
#else
  asm volatile("s_wait_asynccnt 0" ::: "memory");
#endif
}
__device__ __forceinline__ float tanh_f32(float x) {
#if __has_builtin(__builtin_amdgcn_tanhf)
  return __builtin_amdgcn_tanhf(x);          // V_TANH_F32
#elif __has_builtin(__builtin_amdgcn_tanh_f32)
  return __builtin_amdgcn_tanh_f32(x);
#else
  return tanhf(x);
#endif
}

// ---- main fused kernel -----------------------------------------------------
__global__ __launch_bounds__(NTHR)
void sobel_loss_main(const float* __restrict__ pred,
                     const float* __restrict__ gt,
                     const float* __restrict__ mask,
                     float* __restrict__ partial,      // NBLK floats (or null)
                     float* __restrict__ out_atomic)   // fallback target
{
  __shared__ float sP[2][LPLANE];
  __shared__ float sG[2][LPLANE];
  __shared__ float wsum[NTHR / 32];

  const int tx  = threadIdx.x, ty = threadIdx.y;
  const int tid = ty * TW + tx;
  const int w0  = blockIdx.x * TW;
  const int h0  = blockIdx.y * TH;
  const int b   = (int)blockIdx.z / NCHUNK;
  const int z0  = ((int)blockIdx.z % NCHUNK) * CHUNK;
  const size_t volbase = (size_t)b * (size_t)(DIM * PLANE);

  // ---- hoisted staging descriptor: planes are staged in increasing z, so
  // the global source is a single pointer advanced one plane per load and the
  // two LDS destinations (double buffer) are loop-invariant. ----------------
  int role = 2;                 // 0 = interior b128, 1 = halo b32, 2 = idle
  bool hwv = false;             // (h,w) part of validity (z part varies)
  float* dp0 = nullptr;         // LDS dst, buffer 0
  float* dp1 = nullptr;         // LDS dst, buffer 1
  const float* src = nullptr;   // global src for plane z0-1; += PLANE per load

  if (tid < 2 * NINT) {
    role = 0;
    const int arr = (tid >= NINT) ? 1 : 0;
    const int s   = tid - arr * NINT;
    const int r   = s >> 3;                  // LDS row 0..LH-1
    const int q   = s & 7;                   // 4-float quad
    const int gh  = h0 - 1 + r;
    hwv = (unsigned)gh < (unsigned)DIM;      // interior cols always in range
    const int lo  = r * LSTRIDE + 4 + q * 4; // 16B-aligned LDS target
    dp0 = (arr ? sG[0] : sP[0]) + lo;
    dp1 = (arr ? sG[1] : sP[1]) + lo;
    src = (arr ? gt : pred) + volbase
        + (ptrdiff_t)(z0 - 1) * PLANE + (ptrdiff_t)gh * DIM + (w0 + q * 4);
  } else if (tid < 2 * NINT + 2 * NHALO) {
    int idx = tid - 2 * NINT;
    const int arr = (idx >= NHALO) ? 1 : 0;
    idx -= arr * NHALO;
    const int r    = idx >> 1;
    const int side = idx & 1;                // 0 = w0-1, 1 = w0+32
    const int gh   = h0 - 1 + r;
    const int gw   = side ? (w0 + TW) : (w0 - 1);
    role = 1;
    hwv = ((unsigned)gh < (unsigned)DIM) && ((unsigned)gw < (unsigned)DIM);
    const int lo = r * LSTRIDE + (side ? (TW + 4) : 3);
    dp0 = (arr ? sG[0] : sP[0]) + lo;
    dp1 = (arr ? sG[1] : sP[1]) + lo;
    src = (arr ? gt : pred) + volbase
        + (ptrdiff_t)(z0 - 1) * PLANE + (ptrdiff_t)gh * DIM + gw;
  }

  auto load_plane = [&](int gz, int buf) {
    if (role == 0) {
      float* dst = buf ? dp1 : dp0;
      if (hwv && (unsigned)gz < (unsigned)DIM)
        async_ld_b128((uint32_t)(uintptr_t)dst, src);
      else
        *(float4*)dst = make_float4(0.f, 0.f, 0.f, 0.f);  // zero padding
      src += PLANE;
    } else if (role == 1) {
      float* dst = buf ? dp1 : dp0;
      if (hwv && (unsigned)gz < (unsigned)DIM)
        async_ld_b32((uint32_t)(uintptr_t)dst, src);
      else
        dst[0] = 0.0f;
      src += PLANE;
    }
  };

  // Fold an LDS plane into separable 2D partials at this thread's (h,w):
  //   A = s_H(s_W x), B = s_H(d_W x), C = d_H(s_W x), ctr = x(center)
  const int aoff = ty * LSTRIDE + tx + 3;    // col tx+3 = gw-1
  const float* const pPbuf[2] = { sP[0] + aoff, sP[1] + aoff };
  const float* const pGbuf[2] = { sG[0] + aoff, sG[1] + aoff };
  auto abc = [&](const float* p, float& A, float& B, float& C, float& ctr) {
    const float x00 = p[0],            x01 = p[1],                x02 = p[2];
    const float x10 = p[LSTRIDE],      x11 = p[LSTRIDE + 1],      x12 = p[LSTRIDE + 2];
    const float x20 = p[2 * LSTRIDE],  x21 = p[2 * LSTRIDE + 1],  x22 = p[2 * LSTRIDE + 2];
    const float sw0 = fmaf(2.0f, x01, x00) + x02;
    const float sw1 = fmaf(2.0f, x11, x10) + x12;
    const float sw2 = fmaf(2.0f, x21, x20) + x22;
    const float dw0 = x02 - x00;
    const float dw1 = x12 - x10;
    const float dw2 = x22 - x20;
    A = fmaf(2.0f, sw1, sw0) + sw2;
    B = fmaf(2.0f, dw1, dw0) + dw2;
    C = sw0 - sw2;
    ctr = x11;
  };

  float Ap[3], Bp[3], Cp[3], Ag[3], Bg[3], Cg[3];
  float cp1, cg1, cp2, cg2, junk;

  // Prologue: planes z0-1 and z0 into the rings; kick off z0+1.
  load_plane(z0 - 1, 0);
  wait_async0();
  __syncthreads();
  abc(pPbuf[0], Ap[0], Bp[0], Cp[0], junk);
  abc(pGbuf[0], Ag[0], Bg[0], Cg[0], junk);
  load_plane(z0, 1);                 // buf1 has no prior readers
  wait_async0();
  __syncthreads();
  abc(pPbuf[1], Ap[1], Bp[1], Cp[1], cp1);
  abc(pGbuf[1], Ag[1], Bg[1], Cg[1], cg1);
  load_plane(z0 + 1, 0);             // buf0 readers fenced by barrier above

  const float* mp = mask + volbase + (size_t)z0 * PLANE
                  + (size_t)(h0 + ty) * DIM + (w0 + tx);
  float sum = 0.0f;

#pragma unroll 6                     // LCM(ring period 3, buffer parity 2)
  for (int z = z0; z < z0 + CHUNK; ++z) {
    const int rb = (z - z0) & 1;    // buffer holding plane z+1
    wait_async0();
    __syncthreads();                // staging of plane z+1 complete everywhere
    if (z + 1 < z0 + CHUNK)
      load_plane(z + 2, rb ^ 1);    // write rb^1 (readers fenced last iter)
    const float m = *mp; mp += PLANE;
    abc(pPbuf[rb], Ap[2], Bp[2], Cp[2], cp2);
    abc(pGbuf[rb], Ag[2], Bg[2], Cg[2], cg2);

    const float gxp = fmaf(2.0f, Bp[1], Bp[0]) + Bp[2];
    const float gyp = fmaf(2.0f, Cp[1], Cp[0]) + Cp[2];
    const float gzp = Ap[0] - Ap[2];
    const float gxg = fmaf(2.0f, Bg[1], Bg[0]) + Bg[2];
    const float gyg = fmaf(2.0f, Cg[1], Cg[0]) + Cg[2];
    const float gzg = Ag[0] - Ag[2];

    const float magp =
        sqrtf(fmaf(gxp, gxp, fmaf(gyp, gyp, fmaf(gzp, gzp, 1e-10f))));
    const float magg =
        sqrtf(fmaf(gxg, gxg, fmaf(gyg, gyg, fmaf(gzg, gzg, 1e-10f))));

    const float dmg = magg - magp;
    const float mge = dmg * dmg * m;          // reference masks mge pre-tanh
    const float dd  = cp1 - cg1;
    const float mse = dd * dd * m;
    sum += mse * (1.0f + tanh_f32(mge));

    // rotate D-direction rings (killed by renaming under unroll 6)
    Ap[0] = Ap[1]; Ap[1] = Ap[2];  Bp[0] = Bp[1]; Bp[1] = Bp[2];
    Cp[0] = Cp[1]; Cp[1] = Cp[2];  Ag[0] = Ag[1]; Ag[1] = Ag[2];
    Bg[0] = Bg[1]; Bg[1] = Bg[2];  Cg[0] = Cg[1]; Cg[1] = Cg[2];
    cp1 = cp2; cg1 = cg2;
  }

  // wave32 reduction, then cross-wave via LDS (deterministic order)
  for (int off = 16; off > 0; off >>= 1)
    sum += __shfl_down(sum, off, 32);
  const int lane = tid & 31, wid = tid >> 5;
  if (lane == 0) wsum[wid] = sum;
  __syncthreads();
  if (tid == 0) {
    float t = 0.0f;
#pragma unroll
    for (int i = 0; i < NTHR / 32; ++i) t += wsum[i];
    const int blin = ((int)blockIdx.z * GY + (int)blockIdx.y) * GX
                     + (int)blockIdx.x;
    if (partial) partial[blin] = t;
    else atomicAdd(out_atomic, t * (float)(1.0 / NTOT));
  }
}

// ---- deterministic finalize ------------------------------------------------
__global__ void sobel_loss_finalize(const float* __restrict__ partial, int n,
                                    float* __restrict__ out) {
  __shared__ float acc[256];
  float s = 0.0f;
  for (int i = threadIdx.x; i < n; i += 256) s += partial[i];
  acc[threadIdx.x] = s;
  __syncthreads();
  for (int off = 128; off > 0; off >>= 1) {
    if ((int)threadIdx.x < off) acc[threadIdx.x] += acc[threadIdx.x + off];
    __syncthreads();
  }
  if (threadIdx.x == 0) out[0] = acc[0] * (float)(1.0 / NTOT);
}

__global__ void zero_out(float* out) {
  if (threadIdx.x == 0 && blockIdx.x == 0) out[0] = 0.0f;
}

// ---- host entry ------------------------------------------------------------
extern "C" void kernel_launch(void* const* d_in, const int* in_sizes, int n_in,
                              void* d_out, int out_size, void* d_ws,
                              size_t ws_size, hipStream_t stream) {
  (void)in_sizes; (void)n_in; (void)out_size;
  const float* pred = (const float*)d_in[0];
  const float* gt   = (const float*)d_in[1];
  const float* mask = (const float*)d_in[2];
  float* out = (float*)d_out;

  dim3 grid(GX, GY, GZ);
  dim3 block(TW, TH, 1);

  if (ws_size >= (size_t)NBLK * sizeof(float)) {
    float* partial = (float*)d_ws;
    sobel_loss_main<<<grid, block, 0, stream>>>(pred, gt, mask, partial,
                                                nullptr);
    sobel_loss_finalize<<<1, 256, 0, stream>>>(partial, NBLK, out);
  } else {
    zero_out<<<1, 32, 0, stream>>>(out);
    sobel_loss_main<<<grid, block, 0, stream>>>(pred, gt, mask, nullptr, out);
  }
}